// AdvancedFuzzyAttention_5806795784852
// MI455X (gfx1250) — compile-verified
//
#include <hip/hip_runtime.h>
#include <hip/hip_bf16.h>
#include <math.h>

// ---------------------------------------------------------------------------
// Problem constants (reference: B=8, S=512, HID=4096, H=8, F=7, D=512)
// ---------------------------------------------------------------------------
#define BB    8
#define SS    512
#define HID   4096
#define HQ    1024          // HID/4 (gate hidden)
#define MPAD  16            // batch rows padded 8 -> 16 for WMMA M dimension

typedef __attribute__((ext_vector_type(2))) float v2f;
typedef __attribute__((ext_vector_type(8))) float v8f;

// ---------------------------------------------------------------------------
// 1) vmean[r][i] = mean_s value[r,s,i]  (rows 8..15 zero-padded)
// ---------------------------------------------------------------------------
__global__ __launch_bounds__(256)
void k_mean_value(const float* __restrict__ value, float* __restrict__ vmean) {
    const int i = blockIdx.x * 256 + threadIdx.x;   // 0..HID-1
    const int r = blockIdx.y;                       // 0..15
    float acc = 0.0f;
    if (r < BB) {
        const float* p = value + (size_t)r * SS * HID + i;
        #pragma unroll 4
        for (int s = 0; s < SS; ++s) acc += p[(size_t)s * HID];
        acc *= (1.0f / (float)SS);
    }
    vmean[r * HID + i] = acc;
}

// ---------------------------------------------------------------------------
// 2) C[16,N] = A[16,K] @ Bm[K,N] + bias   via V_WMMA_F32_16X16X4_F32
//    One wave32 owns one 16-column tile; 8 waves per block.
//    A rows 8..15 are zero on input; epilogue re-zeroes rows >=8 so the
//    result can chain as the next GEMM's A without masking.
// ---------------------------------------------------------------------------
__global__ __launch_bounds__(256)
void k_wmma_gemm16(const float* __restrict__ A, const float* __restrict__ Bm,
                   const float* __restrict__ bias, float* __restrict__ C,
                   int K, int N) {
    const int wave = threadIdx.x >> 5;
    const int lane = threadIdx.x & 31;
    const int tile = blockIdx.x * 8 + wave;
    const int n    = tile * 16 + (lane & 15);
    const int m    = lane & 15;
    const int koff = (lane >> 4) * 2;               // half-wave K split

    const float* Arow = A  + (size_t)m * K + koff;          // A[m][k+koff], [k+koff+1]
    const float* Bcol = Bm + (size_t)koff * N + n;          // Bm[k+koff][n]

    v8f acc = {};
    #pragma unroll 4
    for (int k = 0; k < K; k += 4) {
        v2f a = *(const v2f*)(Arow + k);                    // 8B load, aligned
        v2f b;
        b.x = Bcol[(size_t)k * N];
        b.y = Bcol[(size_t)k * N + N];
        acc = __builtin_amdgcn_wmma_f32_16x16x4_f32(
                  /*neg_a=*/false, a, /*neg_b=*/false, b,
                  /*c_mod=*/(short)0, acc, /*reuse_a=*/false, /*reuse_b=*/false);
    }

    const int   mbase = (lane >> 4) * 8;
    const float bv    = bias[n];
    #pragma unroll
    for (int r = 0; r < 8; ++r) {
        const int mm = mbase + r;
        C[(size_t)mm * N + n] = (mm < BB) ? (acc[r] + bv) : 0.0f;
    }
}

// ---------------------------------------------------------------------------
// 3) gate[b] = sigmoid( dot(relu(hidden[b,:]), gW2) + gb2 )
// ---------------------------------------------------------------------------
__global__ __launch_bounds__(256)
void k_gate(const float* __restrict__ hidden, const float* __restrict__ gW2,
            const float* __restrict__ gb2, float* __restrict__ gate) {
    __shared__ float red[256];
    const int b = blockIdx.x, t = threadIdx.x;
    float p = 0.0f;
    for (int j = t; j < HQ; j += 256) {
        float h = hidden[b * HQ + j];
        p += fmaxf(h, 0.0f) * gW2[j];
    }
    red[t] = p;
    __syncthreads();
    for (int s = 128; s > 0; s >>= 1) {
        if (t < s) red[t] += red[t + s];
        __syncthreads();
    }
    if (t == 0) gate[b] = 1.0f / (1.0f + expf(-(red[0] + gb2[0])));
}

// ---------------------------------------------------------------------------
// 4) out[b,s,:] = LN( out0[b,:]*gate[b] + query[b,s,:] ) * ln_g + ln_b
// ---------------------------------------------------------------------------
__global__ __launch_bounds__(256)
void k_ln(const float* __restrict__ out0, const float* __restrict__ gate,
          const float* __restrict__ query, const float* __restrict__ ln_g,
          const float* __restrict__ ln_b, float* __restrict__ out) {
    __shared__ float rsum[256], rsq[256];
    __shared__ float mu_s, rstd_s;
    const int s = blockIdx.x, b = blockIdx.y, t = threadIdx.x;
    const float g = gate[b];
    const float* q  = query + ((size_t)(b * SS + s)) * HID;
    const float* o0 = out0  + (size_t)b * HID;

    float xs[HID / 256];
    float su = 0.0f, sq = 0.0f;
    #pragma unroll
    for (int j = 0; j < HID / 256; ++j) {
        const int i = t + j * 256;
        const float x = fmaf(o0[i], g, q[i]);
        xs[j] = x;
        su += x;
        sq += x * x;
    }
    rsum[t] = su; rsq[t] = sq;
    __syncthreads();
    for (int st = 128; st > 0; st >>= 1) {
        if (t < st) { rsum[t] += rsum[t + st]; rsq[t] += rsq[t + st]; }
        __syncthreads();
    }
    if (t == 0) {
        const float mu  = rsum[0] * (1.0f / (float)HID);
        const float var = rsq[0] * (1.0f / (float)HID) - mu * mu;
        mu_s   = mu;
        rstd_s = rsqrtf(var + 1e-5f);
    }
    __syncthreads();
    const float mu = mu_s, rstd = rstd_s;
    float* op = out + ((size_t)(b * SS + s)) * HID;
    #pragma unroll
    for (int j = 0; j < HID / 256; ++j) {
        const int i = t + j * 256;
        op[i] = (xs[j] - mu) * rstd * ln_g[i] + ln_b[i];
    }
}

// ---------------------------------------------------------------------------
// 5) attn output: softmax of a constant row is exactly uniform = 1/512
// ---------------------------------------------------------------------------
__global__ __launch_bounds__(256)
void k_fill_attn(float* __restrict__ p) {
    const size_t i = ((size_t)blockIdx.x * 256 + threadIdx.x) * 4;
    float4 v;
    v.x = v.y = v.z = v.w = 1.0f / 512.0f;
    *(float4*)(p + i) = v;
}

// ---------------------------------------------------------------------------
// Launch
// ---------------------------------------------------------------------------
extern "C" void kernel_launch(void* const* d_in, const int* in_sizes, int n_in,
                              void* d_out, int out_size, void* d_ws, size_t ws_size,
                              hipStream_t stream) {
    (void)in_sizes; (void)n_in; (void)out_size; (void)ws_size;

    const float* query = (const float*)d_in[0];
    // d_in[1] key, d_in[3] Wq, d_in[4] bq, d_in[5] Wk, d_in[6] bk,
    // d_in[11] fuzzy_centers, d_in[12] fuzzy_widths: provably dead
    // (uniform softmax makes the score path irrelevant to both outputs).
    const float* value = (const float*)d_in[2];
    const float* Wv    = (const float*)d_in[7];
    const float* bv    = (const float*)d_in[8];
    const float* Wo    = (const float*)d_in[9];
    const float* bo    = (const float*)d_in[10];
    const float* gW1   = (const float*)d_in[13];
    const float* gb1   = (const float*)d_in[14];
    const float* gW2   = (const float*)d_in[15];
    const float* gb2   = (const float*)d_in[16];
    const float* ln_g  = (const float*)d_in[17];
    const float* ln_b  = (const float*)d_in[18];

    float* out  = (float*)d_out;                        // [B,S,HID]
    float* attn = out + (size_t)BB * SS * HID;          // [B,H,S,S]

    float* ws     = (float*)d_ws;
    float* vmean  = ws;                                 // [16, HID]
    float* meanV  = vmean + MPAD * HID;                 // [16, HID]
    float* out0   = meanV + MPAD * HID;                 // [16, HID]
    float* hidden = out0  + MPAD * HID;                 // [16, HQ]
    float* gate   = hidden + MPAD * HQ;                 // [8]

    // 1) mean over sequence of value (padded to 16 rows of zeros)
    k_mean_value<<<dim3(HID / 256, MPAD), 256, 0, stream>>>(value, vmean);

    // 2) meanV = vmean @ Wv + bv      [16,4096] = [16,4096]x[4096,4096]
    k_wmma_gemm16<<<HID / 128, 256, 0, stream>>>(vmean, Wv, bv, meanV, HID, HID);

    // 3) out0 = meanV @ Wo + bo
    k_wmma_gemm16<<<HID / 128, 256, 0, stream>>>(meanV, Wo, bo, out0, HID, HID);

    // 4) hidden = out0 @ gW1 + gb1    [16,1024]
    k_wmma_gemm16<<<HQ / 128, 256, 0, stream>>>(out0, gW1, gb1, hidden, HID, HQ);

    // 5) gate = sigmoid(relu(hidden) @ gW2 + gb2)
    k_gate<<<BB, 256, 0, stream>>>(hidden, gW2, gb2, gate);

    // 6) fused gate * out0 + residual -> LayerNorm -> out
    k_ln<<<dim3(SS, BB), 256, 0, stream>>>(out0, gate, query, ln_g, ln_b, out);

    // 7) attn = constant 1/512 (exact: softmax of constant rows)
    const size_t attn_elems = (size_t)BB * 8 * SS * SS;     // 16,777,216
    k_fill_attn<<<(unsigned)(attn_elems / 4 / 256), 256, 0, stream>>>(attn);
}